// SpatialTransformationLayer_32478542692854
// MI455X (gfx1250) — compile-verified
//
#include <hip/hip_runtime.h>
#include <hip/hip_bf16.h>

// 3D trilinear warp of a 256^3 volume. Memory-bound: ~320 MiB compulsory
// traffic -> ~13.7 us floor @ 23.3 TB/s. Image (64 MiB) stays resident in the
// 192 MB L2; dvf/out are streamed with non-temporal hints so they never evict
// it. dvf is staged into LDS with CDNA5 async global->LDS copies
// (double-buffered, s_wait_asynccnt); next line's image neighborhood is
// warmed with global_prefetch.

#define LINES 4           // z-lines (256 voxels each) per 256-thread block
#define NVOX 256

// ---- CDNA5 async global->LDS copy (per-lane 4 bytes), tracked by ASYNCcnt ----
// th:TH_LOAD_NT: dvf is a 192 MiB single-use stream; keep it out of L2.
__device__ __forceinline__ void async_g2l_b32_nt(unsigned lds_addr, const float* gaddr) {
    asm volatile("global_load_async_to_lds_b32 %0, %1, off th:TH_LOAD_NT"
                 :: "v"(lds_addr), "v"(gaddr)
                 : "memory");
}

template <int N>
__device__ __forceinline__ void wait_asynccnt() {
#if __has_builtin(__builtin_amdgcn_s_wait_asynccnt)
    __builtin_amdgcn_s_wait_asynccnt(N);
    asm volatile("" ::: "memory");   // keep LDS reads below the wait
#else
    asm volatile("s_wait_asynccnt %0" :: "n"(N) : "memory");
#endif
}

__device__ __forceinline__ unsigned lds_off(const void* p) {
    // generic (flat) LDS pointer: low 32 bits == wave-relative LDS byte offset
    return (unsigned)(size_t)p;
}

__global__ __launch_bounds__(256)
void warp3d_trilinear_kernel(const float* __restrict__ image,
                             const float* __restrict__ dvf,
                             float* __restrict__ out) {
    constexpr long long PLANE = (long long)NVOX * NVOX * NVOX; // 16,777,216
    __shared__ float sbuf[2][3][NVOX];   // double buffer: {dy,dx,dz} per line

    const int tid   = threadIdx.x;       // z coordinate
    const int line0 = blockIdx.x * LINES;

    auto issue_line = [&](int l, int b) {
        const long long t = ((long long)(line0 + l) << 8) + tid;
        async_g2l_b32_nt(lds_off(&sbuf[b][0][tid]), dvf + t);              // dy
        async_g2l_b32_nt(lds_off(&sbuf[b][1][tid]), dvf + PLANE + t);      // dx
        async_g2l_b32_nt(lds_off(&sbuf[b][2][tid]), dvf + 2 * PLANE + t);  // dz
    };

    issue_line(0, 0);

#pragma unroll
    for (int l = 0; l < LINES; ++l) {
        if (l + 1 < LINES) {
            issue_line(l + 1, (l + 1) & 1);  // prefetch next line's dvf
            // Warm the gather neighborhood of the next line: taps cluster
            // around the identity position (displacements ~N(0,1)).
            __builtin_prefetch(image + ((long long)(line0 + l + 1) << 8) + tid, 0, 3);
            wait_asynccnt<3>();              // current line's 3 copies done
        } else {
            wait_asynccnt<0>();
        }

        const int b  = l & 1;
        const float dy = sbuf[b][0][tid];
        const float dx = sbuf[b][1][tid];
        const float dz = sbuf[b][2][tid];

        const int L = line0 + l;
        const int y = L >> 8;
        const int x = L & 255;
        const int z = tid;

        const float ny = (float)y + dy;
        const float nx = (float)x + dx;
        const float nz = (float)z + dz;

        const float fy = floorf(ny);
        const float fx = floorf(nx);
        const float fz = floorf(nz);

        const int y0 = min(max((int)fy,     0), 255);
        const int y1 = min(max((int)fy + 1, 0), 255);
        const int x0 = min(max((int)fx,     0), 255);
        const int x1 = min(max((int)fx + 1, 0), 255);
        const int z0 = min(max((int)fz,     0), 255);
        const int z1 = min(max((int)fz + 1, 0), 255);

        // fractions measured against the CLAMPED low corner (matches reference)
        const float yd = ny - (float)y0;
        const float xd = nx - (float)x0;
        const float zd = nz - (float)z0;

        const int r00 = ((y0 << 8) + x0) << 8;
        const int r01 = ((y0 << 8) + x1) << 8;
        const int r10 = ((y1 << 8) + x0) << 8;
        const int r11 = ((y1 << 8) + x1) << 8;

        const float c000 = image[r00 + z0], c001 = image[r00 + z1];
        const float c010 = image[r01 + z0], c011 = image[r01 + z1];
        const float c100 = image[r10 + z0], c101 = image[r10 + z1];
        const float c110 = image[r11 + z0], c111 = image[r11 + z1];

        const float omz = 1.0f - zd, omx = 1.0f - xd, omy = 1.0f - yd;
        const float c00 = c000 * omz + c001 * zd;
        const float c01 = c010 * omz + c011 * zd;
        const float c10 = c100 * omz + c101 * zd;
        const float c11 = c110 * omz + c111 * zd;
        const float c0  = c00 * omx + c01 * xd;
        const float c1  = c10 * omx + c11 * xd;
        const float r   = c0 * omy + c1 * yd;

        const long long t = ((long long)L << 8) + tid;
        __builtin_nontemporal_store(r, out + t);  // don't pollute L2 (image lives there)
    }
}

extern "C" void kernel_launch(void* const* d_in, const int* in_sizes, int n_in,
                              void* d_out, int out_size, void* d_ws, size_t ws_size,
                              hipStream_t stream) {
    const float* image = (const float*)d_in[0];   // [1,1,256,256,256] f32
    const float* dvf   = (const float*)d_in[1];   // [1,3,256,256,256] f32
    float*       out   = (float*)d_out;           // [1,1,256,256,256] f32

    const int total_lines = NVOX * NVOX;          // 65536 z-lines
    dim3 grid(total_lines / LINES);               // 16384 blocks
    dim3 block(NVOX);                             // 256 threads = 8 wave32
    warp3d_trilinear_kernel<<<grid, block, 0, stream>>>(image, dvf, out);
}